// RelativePositionMessage_73418170958215
// MI455X (gfx1250) — compile-verified
//
#include <hip/hip_runtime.h>

typedef __attribute__((ext_vector_type(2))) float v2f;
typedef __attribute__((ext_vector_type(8))) float v8f;

#define DIM   64
#define PITCH 68          // 17 float4s: keeps b128 stores 16B-aligned, conflict-free banks
#define WPB   4           // waves per block
#define TILE  16          // edges per wave

__device__ __forceinline__ float pick8(v8f v, int i) {
  float r = v[0];
  r = (i == 1) ? v[1] : r;
  r = (i == 2) ? v[2] : r;
  r = (i == 3) ? v[3] : r;
  r = (i == 4) ? v[4] : r;
  r = (i == 5) ? v[5] : r;
  r = (i == 6) ? v[6] : r;
  r = (i == 7) ? v[7] : r;
  return r;
}

__global__ __launch_bounds__(WPB * 32) void rpm_wmma_kernel(
    const float* __restrict__ G, const float* __restrict__ K,
    const float* __restrict__ Q, const float* __restrict__ V,
    const int* __restrict__ src, const int* __restrict__ dst,
    float* __restrict__ out, int E) {
  __shared__ float ldsDiff[WPB][TILE * PITCH];
  __shared__ float ldsK[WPB][TILE * PITCH];
  __shared__ float ldsQ[WPB][TILE * PITCH];

  const int lane = threadIdx.x & 31;
  const int wave = threadIdx.x >> 5;
  const int half = lane >> 4;          // 0: lanes 0-15, 1: lanes 16-31
  const int qcol = (lane & 15) << 2;   // float4 column within a 64-float row

  long tile = (long)blockIdx.x * WPB + wave;
  long e0 = tile * TILE;
  if (e0 + TILE > (long)E) e0 = (long)E - TILE;  // clamp: duplicates rewrite identical data

  float* dW = ldsDiff[wave];
  float* kW = ldsK[wave];
  float* qW = ldsQ[wave];

  // ---- Stage gathered rows into LDS: diff = G[src]-G[dst], K[src], Q[dst] ----
  // Two rows per iteration (one per half-wave), float4 per lane -> 256B/row coalesced.
#pragma unroll
  for (int i = 0; i < 8; ++i) {
    const int r = 2 * i + half;
    const long e = e0 + r;
    const long s = (long)src[e];
    const long d = (long)dst[e];
    const float4 gs = *(const float4*)(G + s * DIM + qcol);
    const float4 gd = *(const float4*)(G + d * DIM + qcol);
    const float4 kk = *(const float4*)(K + s * DIM + qcol);
    const float4 qq = *(const float4*)(Q + d * DIM + qcol);
    float4 df;
    df.x = gs.x - gd.x; df.y = gs.y - gd.y;
    df.z = gs.z - gd.z; df.w = gs.w - gd.w;
    *(float4*)(dW + r * PITCH + qcol) = df;
    *(float4*)(kW + r * PITCH + qcol) = kk;
    *(float4*)(qW + r * PITCH + qcol) = qq;
  }
  __syncthreads();

  // ---- WMMA reductions over K=64 in 16 chunks of 4:
  //   accKQ = K_tile(16x64) * Q_tile(16x64)^T   (need diagonal = per-edge K.Q)
  //   accDD = Diff_tile     * Diff_tile^T       (need diagonal = per-edge |diff|^2)
  // A layout (16x4 f32): lane m holds A[m][0..1], lane m+16 holds A[m][2..3].
  // B layout (4x16 f32): lane n holds B[0..1][n], lane n+16 holds B[2..3][n] = Q[n][k..].
  // => both operands use identical per-lane addressing: row = lane&15, col = 4c + 2*half.
  v8f accKQ = {};
  v8f accDD = {};
  const int row = lane & 15;
#pragma unroll
  for (int c = 0; c < 16; ++c) {
    const int co = 4 * c + 2 * half;
    const float2 ak = *(const float2*)(kW + row * PITCH + co);
    const float2 bq = *(const float2*)(qW + row * PITCH + co);
    const float2 ad = *(const float2*)(dW + row * PITCH + co);
    v2f a  = {ak.x, ak.y};
    v2f b  = {bq.x, bq.y};
    v2f d2 = {ad.x, ad.y};
    accKQ = __builtin_amdgcn_wmma_f32_16x16x4_f32(
        false, a, false, b, (short)0, accKQ, false, false);
    accDD = __builtin_amdgcn_wmma_f32_16x16x4_f32(
        false, d2, false, d2, (short)0, accDD, false, false);
  }

  // ---- Diagonal extraction + weight.
  // C/D layout: vgpr v, lanes 0-15 -> (M=v, N=lane); lanes 16-31 -> (M=v+8, N=lane-16).
  // Diag i<8 : lane i,    vgpr i.    Diag i>=8: lane i+16, vgpr i-8.
  const float inv = 0.125f;  // 1/sqrt(64)
  const int idx = lane & 7;
  const float kq = pick8(accKQ, idx);
  const float dd = pick8(accDD, idx);
  const float sc = fminf(fmaxf(kq * inv, -5.0f), 5.0f);
  const float di = fminf(fmaxf(-__builtin_sqrtf(dd + 1e-6f) * inv, -5.0f), 5.0f);
  const float w = __expf(sc + di);  // exp(clip(score))*exp(clip(dist))

  // ---- Output: out[e] = w[e] * V[src[e]]; weight of edge r lives on lane r (r<8) or r+16.
#pragma unroll
  for (int i = 0; i < 8; ++i) {
    const int r = 2 * i + half;
    const int srcLane = (r < 8) ? r : (r + 16);
    const float wr = __shfl(w, srcLane, 32);
    const long e = e0 + r;
    const long s = (long)src[e];
    const float4 vv = *(const float4*)(V + s * DIM + qcol);
    float4 o;
    o.x = wr * vv.x; o.y = wr * vv.y;
    o.z = wr * vv.z; o.w = wr * vv.w;
    *(float4*)(out + e * DIM + qcol) = o;
  }
}

extern "C" void kernel_launch(void* const* d_in, const int* in_sizes, int n_in,
                              void* d_out, int out_size, void* d_ws, size_t ws_size,
                              hipStream_t stream) {
  const float* G = (const float*)d_in[0];
  const float* K = (const float*)d_in[1];
  const float* Q = (const float*)d_in[2];
  const float* V = (const float*)d_in[3];
  const int* src = (const int*)d_in[4];
  const int* dst = (const int*)d_in[5];
  float* out = (float*)d_out;

  const int E = in_sizes[4];
  if (E < TILE) return;  // not expected (E = 1M)

  const int nTiles = (E + TILE - 1) / TILE;
  const int nBlocks = (nTiles + WPB - 1) / WPB;
  rpm_wmma_kernel<<<nBlocks, WPB * 32, 0, stream>>>(G, K, Q, V, src, dst, out, E);
}